// DGLDigitCapsuleLayer_38783554683542
// MI455X (gfx1250) — compile-verified
//
#include <hip/hip_runtime.h>
#include <hip/hip_bf16.h>
#include <math.h>

// Problem constants (match reference)
#define NIN   1152
#define NOUT  10
#define DIN   8
#define DOUT  16
#define BATCH 256
#define KTOT  (NIN * DIN)   // 9216
#define EPSF  1e-9f

typedef float v2f __attribute__((ext_vector_type(2)));
typedef float v8f __attribute__((ext_vector_type(8)));

// ---------------------------------------------------------------------------
// prep: X2[b, i, d] = x[b, d, i]  (contiguous (i,d) K-axis for WMMA A loads)
//       also zero the routing logits bb[NIN*NOUT]
// ---------------------------------------------------------------------------
__global__ __launch_bounds__(256) void prep_kernel(const float* __restrict__ x,
                                                   float* __restrict__ X2,
                                                   float* __restrict__ bb) {
  size_t t = (size_t)blockIdx.x * blockDim.x + threadIdx.x;
  if (t < (size_t)NIN * NOUT) bb[t] = 0.0f;
  if (t < (size_t)BATCH * KTOT) {
    int b   = (int)(t / KTOT);
    int rem = (int)(t % KTOT);
    int i   = rem / DIN;
    int d   = rem % DIN;
    X2[t] = x[(size_t)b * KTOT + (size_t)d * NIN + i];
  }
}

// ---------------------------------------------------------------------------
// softmax over j (10) of routing logits -> coupling coefficients c
// ---------------------------------------------------------------------------
__global__ __launch_bounds__(128) void softmax_kernel(const float* __restrict__ bb,
                                                      float* __restrict__ c) {
  int i = blockIdx.x * blockDim.x + threadIdx.x;
  if (i >= NIN) return;
  float vals[NOUT];
  float m = -3.0e38f;
#pragma unroll
  for (int j = 0; j < NOUT; ++j) {
    vals[j] = bb[i * NOUT + j];
    m = fmaxf(m, vals[j]);
  }
  float sum = 0.0f;
#pragma unroll
  for (int j = 0; j < NOUT; ++j) {
    vals[j] = __expf(vals[j] - m);
    sum += vals[j];
  }
  float inv = 1.0f / sum;
#pragma unroll
  for (int j = 0; j < NOUT; ++j) c[i * NOUT + j] = vals[j] * inv;
}

// ---------------------------------------------------------------------------
// s_squash: for one (j, 16-row batch tile):
//   s[b,o] = sum_{i,d} (c[i,j]*W[i,j,o,d]) * X2[b,i,d]   via fp32 WMMA 16x16x4
//   K = 9216 split across 8 waves (i-chunks of 144), LDS reduce, fused squash.
// Output v written to d_out layout [B, NOUT, DOUT].
// ---------------------------------------------------------------------------
__global__ __launch_bounds__(256) void s_squash_kernel(const float* __restrict__ X2,
                                                       const float* __restrict__ W,
                                                       const float* __restrict__ c,
                                                       float* __restrict__ v) {
  const int blk  = blockIdx.x;        // 160 blocks: j fastest
  const int j    = blk % NOUT;
  const int b0   = (blk / NOUT) * 16; // batch tile origin
  const int wave = threadIdx.x >> 5;
  const int lane = threadIdx.x & 31;
  const int lh   = lane >> 4;         // 0/1 half-wave
  const int ll   = lane & 15;

  v8f acc = {};
  const int i0 = wave * (NIN / 8);    // 144 capsules per wave
  const float* xrow = X2 + (size_t)(b0 + ll) * KTOT;

  for (int i = i0; i < i0 + NIN / 8; ++i) {
    const float cij = c[i * NOUT + j];
    const float* wrow = W + ((size_t)(i * NOUT + j) * DOUT + ll) * DIN;
#pragma unroll
    for (int d0 = 0; d0 < DIN; d0 += 4) {
      // A fragment: A[m=ll, k = d0+2*lh+{0,1}] -> one b64 load
      v2f a = *(const v2f*)(xrow + (size_t)i * DIN + d0 + 2 * lh);
      // B fragment: B[k = d0+2*lh+{0,1}, n=ll] = c * W[i,j,o=ll,d] -> one b64 load
      v2f bm = *(const v2f*)(wrow + d0 + 2 * lh);
      bm[0] *= cij;
      bm[1] *= cij;
      acc = __builtin_amdgcn_wmma_f32_16x16x4_f32(false, a, false, bm,
                                                  (short)0, acc, false, false);
    }
  }

  // Cross-wave reduction of the 16x16 tile in LDS
  __shared__ float red[8][256];
  __shared__ float sbuf[256];
  __shared__ float scale[16];
#pragma unroll
  for (int e = 0; e < 8; ++e) {
    int m = e + 8 * lh;                    // C/D layout: M = e + 8*(lane>=16)
    red[wave][m * 16 + ll] = acc[e];
  }
  __syncthreads();
  {
    float ssum = 0.0f;
#pragma unroll
    for (int wv = 0; wv < 8; ++wv) ssum += red[wv][threadIdx.x];
    sbuf[threadIdx.x] = ssum;              // s for (b_local = t/16, o = t%16)
  }
  __syncthreads();
  if (threadIdx.x < 16) {
    float sq = 0.0f;
#pragma unroll
    for (int o = 0; o < DOUT; ++o) {
      float t = sbuf[threadIdx.x * 16 + o];
      sq += t * t;
    }
    scale[threadIdx.x] = (sq / (1.0f + sq)) * rsqrtf(sq + EPSF);
  }
  __syncthreads();
  {
    int m = threadIdx.x >> 4, o = threadIdx.x & 15;
    v[(size_t)(b0 + m) * (NOUT * DOUT) + j * DOUT + o] = sbuf[threadIdx.x] * scale[m];
  }
}

// ---------------------------------------------------------------------------
// gmat: G[j, k, o] = sum_b X2[b, k] * v[b, j, o]  (K-reduction over batch=256)
// one wave per (j, 16-wide k tile): 64 fp32 WMMAs.
// ---------------------------------------------------------------------------
__global__ __launch_bounds__(32) void gmat_kernel(const float* __restrict__ X2,
                                                  const float* __restrict__ v,
                                                  float* __restrict__ G) {
  const int blk = blockIdx.x;          // 10 * 576 blocks, j fastest
  const int j   = blk % NOUT;
  const int k0  = (blk / NOUT) * 16;
  const int lane = threadIdx.x & 31;
  const int lh = lane >> 4, ll = lane & 15;

  v8f acc = {};
  for (int bb = 0; bb < BATCH; bb += 4) {
    v2f a, bm;
    // A[m = k-local = ll, kk = 2*lh + r] = X2[bb+kk, k0+ll]
    a[0] = X2[(size_t)(bb + 2 * lh + 0) * KTOT + k0 + ll];
    a[1] = X2[(size_t)(bb + 2 * lh + 1) * KTOT + k0 + ll];
    // B[kk = 2*lh + r, n = o = ll] = v[bb+kk, j, ll]
    bm[0] = v[(size_t)(bb + 2 * lh + 0) * (NOUT * DOUT) + j * DOUT + ll];
    bm[1] = v[(size_t)(bb + 2 * lh + 1) * (NOUT * DOUT) + j * DOUT + ll];
    acc = __builtin_amdgcn_wmma_f32_16x16x4_f32(false, a, false, bm,
                                                (short)0, acc, false, false);
  }
#pragma unroll
  for (int e = 0; e < 8; ++e) {
    int m = e + 8 * lh;                  // k-local
    G[((size_t)j * KTOT + k0 + m) * DOUT + ll] = acc[e];
  }
}

// ---------------------------------------------------------------------------
// bupdate: b[i,j] += (1/B) * sum_{o,d} W[i,j,o,d] * G[j, i*8+d, o]
// ---------------------------------------------------------------------------
__global__ __launch_bounds__(128) void bupdate_kernel(const float* __restrict__ W,
                                                      const float* __restrict__ G,
                                                      float* __restrict__ bb) {
  int t = blockIdx.x * blockDim.x + threadIdx.x;
  if (t >= NIN * NOUT) return;
  int i = t / NOUT, j = t % NOUT;
  const float* wrow = W + (size_t)t * (DOUT * DIN);                 // [o][d]
  const float* grow = G + ((size_t)j * KTOT + (size_t)i * DIN) * DOUT; // [d][o]
  float acc = 0.0f;
#pragma unroll
  for (int o = 0; o < DOUT; ++o)
#pragma unroll
    for (int d = 0; d < DIN; ++d)
      acc += wrow[o * DIN + d] * grow[d * DOUT + o];
  bb[t] += acc * (1.0f / BATCH);
}

// ---------------------------------------------------------------------------
// launch: workspace layout (floats):
//   X2 [BATCH*KTOT]=2359296 | G [NOUT*KTOT*DOUT]=1474560 | bb [11520] | c [11520]
//   total 3,856,896 floats = 15.43 MB
// ---------------------------------------------------------------------------
extern "C" void kernel_launch(void* const* d_in, const int* in_sizes, int n_in,
                              void* d_out, int out_size, void* d_ws, size_t ws_size,
                              hipStream_t stream) {
  const float* x = (const float*)d_in[0];   // [256, 8, 1152]
  const float* W = (const float*)d_in[1];   // [1152, 10, 16, 8]
  float* out = (float*)d_out;               // [256, 10, 16] == v
  float* ws  = (float*)d_ws;

  float* X2 = ws;                                   // [BATCH, KTOT]
  float* G  = X2 + (size_t)BATCH * KTOT;            // [NOUT, KTOT, DOUT]
  float* bb = G + (size_t)NOUT * KTOT * DOUT;       // [NIN, NOUT] routing logits
  float* c  = bb + (size_t)NIN * NOUT;              // [NIN, NOUT] couplings

  prep_kernel<<<(BATCH * KTOT + 255) / 256, 256, 0, stream>>>(x, X2, bb);

  for (int it = 0; it < 3; ++it) {
    softmax_kernel<<<(NIN + 127) / 128, 128, 0, stream>>>(bb, c);
    // writes v directly into d_out (also consumed by agreement step)
    s_squash_kernel<<<NOUT * (BATCH / 16), 256, 0, stream>>>(X2, W, c, out);
    if (it < 2) {
      gmat_kernel<<<NOUT * (KTOT / 16), 32, 0, stream>>>(X2, out, G);
      bupdate_kernel<<<(NIN * NOUT + 127) / 128, 128, 0, stream>>>(W, G, bb);
    }
  }
}